// Transformer_Dec_88957362635397
// MI455X (gfx1250) — compile-verified
//
#include <hip/hip_runtime.h>
#include <hip/hip_bf16.h>

// ---------------------------------------------------------------------------
// Deformable-DETR decoder (3 layers) for MI455X / gfx1250, fp32 end-to-end.
// All linear layers run on the CDNA5 matrix pipe via V_WMMA_F32_16X16X4_F32.
// GEMM inner loops are branch-free (clamped row pointers, unconditional b64
// loads); epilogues use a wave-uniform full-tile fast path so interior tiles
// store with zero divergent branches.
// ---------------------------------------------------------------------------

typedef float v2f __attribute__((ext_vector_type(2)));
typedef float v8f __attribute__((ext_vector_type(8)));

constexpr int C_    = 256;
constexpr int H_    = 8;
constexpr int D_    = 32;
constexpr int L_    = 3;
constexpr int P_    = 3;
constexpr int T_    = 3;
constexpr int B_    = 2;
constexpr int Q_    = 300;
constexpr int BT_   = 6;          // B*T
constexpr int DFF_  = 1024;
constexpr int NSRC_ = 19320;      // sum Hl*Wl
constexpr int CT_   = 1;          // (T-1)/2
constexpr int M1_   = BT_ * Q_;   // 1800 rows (video queries)
constexpr int M2_   = B_ * Q_;    // 600 rows (instance queries)

enum { ACT_NONE = 0, ACT_RELU = 1, ACT_GELU = 2 };

template <int ACT>
__device__ __forceinline__ float apply_act(float v)
{
  if (ACT == ACT_RELU) v = fmaxf(v, 0.f);
  if (ACT == ACT_GELU) v = 0.5f * v * (1.f + erff(v * 0.70710678118f));
  return v;
}

// ---------------------------------------------------------------------------
// Generic GEMM: Y[M,N] = act(X[M,K] @ W[N,K]^T + bias[N]).  K % 4 == 0.
// One wave32 computes a 32(M) x 64(N) block: 2 row-tiles x 4 col-tiles = 8
// independent WMMA accumulation chains (hides WMMA RAW latency); per K-step
// 6 b64 loads feed 8 WMMAs.
// A frag (16x4 f32): lanes 0-15 = rows M, half-wave selects K pair {0,1}/{2,3}.
// B frag (4x16):     lanes = cols N, same K split.
// C/D (16x16 f32):   VGPR r, lanes 0-15 -> M=r, lanes 16-31 -> M=8+r, N=lane&15.
// ---------------------------------------------------------------------------
template <int ACT>
__global__ __launch_bounds__(256) void gemm_bias_kernel(
    const float* __restrict__ X, const float* __restrict__ W,
    const float* __restrict__ bias, float* __restrict__ Y,
    int M, int N, int K)
{
  const int tilesM2 = (M + 31) >> 5;
  const int tilesN4 = (N + 63) >> 6;
  const int wave    = blockIdx.x * (blockDim.x >> 5) + (threadIdx.x >> 5);
  if (wave >= tilesM2 * tilesN4) return;
  const int tm   = wave / tilesN4;
  const int tn4  = wave % tilesN4;
  const int lane = threadIdx.x & 31;
  const int half = lane >> 4;
  const int l16  = lane & 15;

  const int arow0 = tm * 32 + l16;
  const int arow1 = arow0 + 16;
  const float* __restrict__ xr0 = X + (size_t)(arow0 < M ? arow0 : M - 1) * K;
  const float* __restrict__ xr1 = X + (size_t)(arow1 < M ? arow1 : M - 1) * K;

  int bc[4];
  const float* __restrict__ wr[4];
#pragma unroll
  for (int j = 0; j < 4; ++j) {
    bc[j] = tn4 * 64 + j * 16 + l16;
    wr[j] = W + (size_t)(bc[j] < N ? bc[j] : N - 1) * K;
  }

  v8f acc0[4] = {};
  v8f acc1[4] = {};
#pragma unroll 2
  for (int k0 = 0; k0 < K; k0 += 4) {
    const int ka = k0 + half * 2;
    const v2f a0 = *(const v2f*)(xr0 + ka);
    const v2f a1 = *(const v2f*)(xr1 + ka);
#pragma unroll
    for (int j = 0; j < 4; ++j) {
      const v2f b = *(const v2f*)(wr[j] + ka);
      acc0[j] = __builtin_amdgcn_wmma_f32_16x16x4_f32(false, a0, false, b,
                                                      (short)0, acc0[j],
                                                      false, false);
      acc1[j] = __builtin_amdgcn_wmma_f32_16x16x4_f32(false, a1, false, b,
                                                      (short)0, acc1[j],
                                                      false, false);
    }
  }

  const bool fullM = (tm * 32 + 32) <= M;   // wave-uniform
  const bool fullN = (tn4 * 64 + 64) <= N;  // wave-uniform
  if (fullM && fullN) {
#pragma unroll
    for (int j = 0; j < 4; ++j) {
      const float bv = bias ? bias[bc[j]] : 0.f;
      float* y0 = Y + (size_t)(tm * 32 + half * 8) * N + bc[j];
      float* y1 = y0 + (size_t)16 * N;
#pragma unroll
      for (int r = 0; r < 8; ++r) {
        y0[(size_t)r * N] = apply_act<ACT>(acc0[j][r] + bv);
        y1[(size_t)r * N] = apply_act<ACT>(acc1[j][r] + bv);
      }
    }
  } else {
#pragma unroll
    for (int j = 0; j < 4; ++j) {
      if (bc[j] >= N) continue;
      const float bv = bias ? bias[bc[j]] : 0.f;
#pragma unroll
      for (int r = 0; r < 8; ++r) {
        const int crow0 = tm * 32 + half * 8 + r;
        const int crow1 = crow0 + 16;
        if (crow0 < M) Y[(size_t)crow0 * N + bc[j]] = apply_act<ACT>(acc0[j][r] + bv);
        if (crow1 < M) Y[(size_t)crow1 * N + bc[j]] = apply_act<ACT>(acc1[j][r] + bv);
      }
    }
  }
}

// ---------------------------------------------------------------------------
// Batched attention scores: S[b,h,i,j] = scale * sum_d Qh[b,i,h,d]*Kh[b,j,h,d]
// Qh/Kh stored as (Bn*Qn, C) with head h at column offset h*D.  WMMA, K=D=32.
// ---------------------------------------------------------------------------
__global__ __launch_bounds__(256) void attn_scores_kernel(
    const float* __restrict__ Qh, const float* __restrict__ Kh,
    float* __restrict__ S, int Bn, int Qn, float scale)
{
  const int tiles  = (Qn + 15) >> 4;
  const int per_bh = tiles * tiles;
  const int wave   = blockIdx.x * (blockDim.x >> 5) + (threadIdx.x >> 5);
  if (wave >= Bn * H_ * per_bh) return;
  const int bh = wave / per_bh;
  const int t  = wave % per_bh;
  const int b  = bh / H_;
  const int h  = bh % H_;
  const int ti = t / tiles;
  const int tj = t % tiles;
  const int lane = threadIdx.x & 31;
  const int half = lane >> 4;
  const int l16  = lane & 15;
  const int arow = ti * 16 + l16;
  const int bcol = tj * 16 + l16;
  const float* qb = Qh + (size_t)(b * Qn + (arow < Qn ? arow : Qn - 1)) * C_ + h * D_;
  const float* kb = Kh + (size_t)(b * Qn + (bcol < Qn ? bcol : Qn - 1)) * C_ + h * D_;

  v8f c = {};
#pragma unroll
  for (int k0 = 0; k0 < D_; k0 += 4) {
    const int ka = k0 + half * 2;
    const v2f a  = *(const v2f*)(qb + ka);
    const v2f bb = *(const v2f*)(kb + ka);
    c = __builtin_amdgcn_wmma_f32_16x16x4_f32(false, a, false, bb,
                                              (short)0, c, false, false);
  }
  float* srow = S + (size_t)(b * H_ + h) * Qn * Qn;
  const bool full = (ti * 16 + 16) <= Qn && (tj * 16 + 16) <= Qn;  // uniform
  if (full) {
    float* y = srow + (size_t)(ti * 16 + half * 8) * Qn + bcol;
#pragma unroll
    for (int r = 0; r < 8; ++r) y[(size_t)r * Qn] = c[r] * scale;
  } else {
    if (bcol >= Qn) return;
#pragma unroll
    for (int r = 0; r < 8; ++r) {
      const int crow = ti * 16 + half * 8 + r;
      if (crow < Qn) srow[(size_t)crow * Qn + bcol] = c[r] * scale;
    }
  }
}

// ---------------------------------------------------------------------------
// O[b,i,h*D+d] = sum_j S[b,h,i,j] * Vh[b,j,h*D+d].  WMMA, K = Qn (mult of 4).
// One wave covers 16 rows x 32 cols (both D tiles) -> 2 accumulator chains.
// ---------------------------------------------------------------------------
__global__ __launch_bounds__(256) void attn_apply_kernel(
    const float* __restrict__ S, const float* __restrict__ Vh,
    float* __restrict__ O, int Bn, int Qn)
{
  const int tiles = (Qn + 15) >> 4;  // row tiles
  const int wave  = blockIdx.x * (blockDim.x >> 5) + (threadIdx.x >> 5);
  if (wave >= Bn * H_ * tiles) return;
  const int bh = wave / tiles;
  const int ti = wave % tiles;
  const int b  = bh / H_;
  const int h  = bh % H_;
  const int lane = threadIdx.x & 31;
  const int half = lane >> 4;
  const int l16  = lane & 15;
  const int arow = ti * 16 + l16;
  const float* sb = S + (size_t)(b * H_ + h) * Qn * Qn +
                    (size_t)(arow < Qn ? arow : Qn - 1) * Qn;
  const float* vb0 = Vh + (size_t)b * Qn * C_ + h * D_ + l16;
  const float* vb1 = vb0 + 16;

  v8f c0 = {}, c1 = {};
  for (int k0 = 0; k0 < Qn; k0 += 4) {
    const int ka = k0 + half * 2;
    const v2f a = *(const v2f*)(sb + ka);
    v2f b0, b1;
    b0.x = vb0[(size_t)ka * C_];
    b0.y = vb0[(size_t)(ka + 1) * C_];
    b1.x = vb1[(size_t)ka * C_];
    b1.y = vb1[(size_t)(ka + 1) * C_];
    c0 = __builtin_amdgcn_wmma_f32_16x16x4_f32(false, a, false, b0,
                                               (short)0, c0, false, false);
    c1 = __builtin_amdgcn_wmma_f32_16x16x4_f32(false, a, false, b1,
                                               (short)0, c1, false, false);
  }
  const bool full = (ti * 16 + 16) <= Qn;  // wave-uniform
  if (full) {
    float* y = O + (size_t)(b * Qn + ti * 16 + half * 8) * C_ + h * D_ + l16;
#pragma unroll
    for (int r = 0; r < 8; ++r) {
      y[(size_t)r * C_]      = c0[r];
      y[(size_t)r * C_ + 16] = c1[r];
    }
  } else {
#pragma unroll
    for (int r = 0; r < 8; ++r) {
      const int crow = ti * 16 + half * 8 + r;
      if (crow < Qn) {
        float* orow = O + (size_t)(b * Qn + crow) * C_ + h * D_ + l16;
        orow[0]  = c0[r];
        orow[16] = c1[r];
      }
    }
  }
}

// ---------------------------------------------------------------------------
// LayerNorm over C=256: one wave per row, lane owns 8 strided elements.
// Y = LN(A (+ Bres)) * g + be.  Safe in-place (Y may alias A).
// ---------------------------------------------------------------------------
__global__ __launch_bounds__(256) void ln_kernel(
    const float* __restrict__ A, const float* __restrict__ Bres,
    const float* __restrict__ g, const float* __restrict__ be,
    float* __restrict__ Y, int M)
{
  const int wave = blockIdx.x * (blockDim.x >> 5) + (threadIdx.x >> 5);
  if (wave >= M) return;
  const int lane = threadIdx.x & 31;
  const float* ar = A + (size_t)wave * C_;
  const float* br = Bres ? Bres + (size_t)wave * C_ : nullptr;
  float v[8];
  float s = 0.f;
#pragma unroll
  for (int i = 0; i < 8; ++i) {
    const int c = lane + i * 32;
    v[i] = ar[c] + (br ? br[c] : 0.f);
    s += v[i];
  }
#pragma unroll
  for (int o = 16; o > 0; o >>= 1) s += __shfl_xor(s, o, 32);
  const float mu = s * (1.f / 256.f);
  float var = 0.f;
#pragma unroll
  for (int i = 0; i < 8; ++i) { const float d = v[i] - mu; var += d * d; }
#pragma unroll
  for (int o = 16; o > 0; o >>= 1) var += __shfl_xor(var, o, 32);
  const float rs = rsqrtf(var * (1.f / 256.f) + 1e-5f);
  float* yr = Y + (size_t)wave * C_;
#pragma unroll
  for (int i = 0; i < 8; ++i) {
    const int c = lane + i * 32;
    yr[c] = (v[i] - mu) * rs * g[c] + be[c];
  }
}

// Wave-per-row softmax over n elements, in place.
__global__ __launch_bounds__(256) void softmax_rows_kernel(
    float* __restrict__ S, int rows, int n)
{
  const int wave = blockIdx.x * (blockDim.x >> 5) + (threadIdx.x >> 5);
  if (wave >= rows) return;
  const int lane = threadIdx.x & 31;
  float* r = S + (size_t)wave * n;
  float m = -3.4e38f;
  for (int j = lane; j < n; j += 32) m = fmaxf(m, r[j]);
#pragma unroll
  for (int o = 16; o > 0; o >>= 1) m = fmaxf(m, __shfl_xor(m, o, 32));
  float s = 0.f;
  for (int j = lane; j < n; j += 32) { const float e = __expf(r[j] - m); r[j] = e; s += e; }
#pragma unroll
  for (int o = 16; o > 0; o >>= 1) s += __shfl_xor(s, o, 32);
  const float inv = 1.f / s;
  for (int j = lane; j < n; j += 32) r[j] *= inv;
}

// ---------------------------------------------------------------------------
// Multiscale deformable attention sampling.  One wave per (b,q,h); lane = d
// (D=32 exactly one wave).  9-way attention softmax in registers.
// ---------------------------------------------------------------------------
__global__ __launch_bounds__(256) void msdeform_kernel(
    const float* __restrict__ val, const float* __restrict__ off,
    const float* __restrict__ attl, const float* __restrict__ ref,
    float* __restrict__ out)
{
  const int wave = blockIdx.x * (blockDim.x >> 5) + (threadIdx.x >> 5);
  if (wave >= BT_ * Q_ * H_) return;
  const int h  = wave % H_;
  const int bq = wave / H_;
  const int b  = bq / Q_;
  const int d  = threadIdx.x & 31;

  const float* al = attl + (size_t)bq * (H_ * L_ * P_) + h * (L_ * P_);
  float aw[9];
  float m = -3.4e38f;
#pragma unroll
  for (int i = 0; i < 9; ++i) { aw[i] = al[i]; m = fmaxf(m, aw[i]); }
  float s = 0.f;
#pragma unroll
  for (int i = 0; i < 9; ++i) { aw[i] = __expf(aw[i] - m); s += aw[i]; }
  const float inv = 1.f / s;

  const float* of = off + (size_t)bq * (H_ * L_ * P_ * 2) + h * (L_ * P_ * 2);
  const float* rb = ref + (size_t)bq * 4;
  const float cx = rb[0], cy = rb[1], bw = rb[2], bh = rb[3];
  const float* vbase = val + (size_t)b * NSRC_ * C_ + h * D_ + d;

  const int HLt[3] = {92, 46, 23};
  const int WLt[3] = {160, 80, 40};
  const int STt[3] = {0, 14720, 18400};

  float acc = 0.f;
#pragma unroll
  for (int l = 0; l < 3; ++l) {
    const int Hl = HLt[l], Wl = WLt[l];
    const float* vl = vbase + (size_t)STt[l] * C_;
#pragma unroll
    for (int p = 0; p < 3; ++p) {
      const int ip = l * 3 + p;
      const float lx = cx + of[ip * 2 + 0] * (1.f / 3.f) * bw * 0.5f;
      const float ly = cy + of[ip * 2 + 1] * (1.f / 3.f) * bh * 0.5f;
      const float xf = lx * (float)Wl - 0.5f;
      const float yf = ly * (float)Hl - 0.5f;
      const float x0 = floorf(xf), y0 = floorf(yf);
      const float wx1 = xf - x0, wy1 = yf - y0;
      float smp = 0.f;
#pragma unroll
      for (int dy = 0; dy < 2; ++dy) {
#pragma unroll
        for (int dx = 0; dx < 2; ++dx) {
          const float xi = x0 + (float)dx, yi = y0 + (float)dy;
          const bool valid = (xi >= 0.f) && (xi < (float)Wl) &&
                             (yi >= 0.f) && (yi < (float)Hl);
          if (valid) {
            const float w = (dx ? wx1 : 1.f - wx1) * (dy ? wy1 : 1.f - wy1);
            const int idx = (int)yi * Wl + (int)xi;
            smp += w * vl[(size_t)idx * C_];
          }
        }
      }
      acc += aw[ip] * inv * smp;
    }
  }
  out[(size_t)bq * C_ + h * D_ + d] = acc;
}

// ----------------------------- elementwise ---------------------------------
__global__ __launch_bounds__(256) void add_kernel(
    const float* __restrict__ a, const float* __restrict__ b,
    float* __restrict__ y, int n)
{
  const int i = blockIdx.x * blockDim.x + threadIdx.x;
  if (i < n) y[i] = a[i] + b[i];
}

__global__ __launch_bounds__(256) void sigboxes_kernel(
    const float* __restrict__ bb, const float* __restrict__ ref,
    float* __restrict__ out, int n)
{
  const int i = blockIdx.x * blockDim.x + threadIdx.x;
  if (i >= n) return;
  float r = fminf(fmaxf(ref[i], 0.f), 1.f);
  const float iv = logf(fmaxf(r, 1e-5f) / fmaxf(1.f - r, 1e-5f));
  const float z = bb[i] + iv;
  out[i] = 1.f / (1.f + __expf(-z));
}

// Rows of length 2 -> rows of length 4 padded with `padv`.
// Used for (a) building ref0 from ref points (padv=0.1), and (b) widening the
// p2p level-0 weight buffer to K=4 (pad contributes 0 because the
// matching operand columns are zero).
__global__ __launch_bounds__(256) void pad2to4_kernel(
    const float* __restrict__ src2, float* __restrict__ dst4, float padv, int M)
{
  const int i = blockIdx.x * blockDim.x + threadIdx.x;
  if (i >= M) return;
  dst4[i * 4 + 0] = src2[i * 2 + 0];
  dst4[i * 4 + 1] = src2[i * 2 + 1];
  dst4[i * 4 + 2] = padv;
  dst4[i * 4 + 3] = padv;
}

// Extract box centers into a stride-4, zero-padded buffer (K=4 GEMM input).
__global__ __launch_bounds__(256) void slice2pad4_kernel(
    const float* __restrict__ boxes, float* __restrict__ pts4, int M)
{
  const int i = blockIdx.x * blockDim.x + threadIdx.x;
  if (i >= M) return;
  pts4[i * 4 + 0] = boxes[i * 4 + 0];
  pts4[i * 4 + 1] = boxes[i * 4 + 1];
  pts4[i * 4 + 2] = 0.f;
  pts4[i * 4 + 3] = 0.f;
}

__global__ __launch_bounds__(256) void instboxes_kernel(
    const float* __restrict__ ref, float* __restrict__ iref)
{
  const int i = blockIdx.x * blockDim.x + threadIdx.x;
  if (i >= B_ * Q_) return;
  const int b = i / Q_, q = i % Q_;
  float lox = 1e30f, loy = 1e30f, hix = -1e30f, hiy = -1e30f;
  for (int t = 0; t < T_; ++t) {
    const float* r = ref + ((size_t)(b * T_ + t) * Q_ + q) * 4;
    float x0 = fminf(fmaxf(r[0] - 0.5f * r[2], 0.f), 1.f);
    float y0 = fminf(fmaxf(r[1] - 0.5f * r[3], 0.f), 1.f);
    float x1 = fminf(fmaxf(r[0] + 0.5f * r[2], 0.f), 1.f);
    float y1 = fminf(fmaxf(r[1] + 0.5f * r[3], 0.f), 1.f);
    lox = fminf(lox, x0); loy = fminf(loy, y0);
    hix = fmaxf(hix, x1); hiy = fmaxf(hiy, y1);
  }
  float* o = iref + (size_t)i * 4;
  o[0] = (lox + hix) * 0.5f;
  o[1] = (loy + hiy) * 0.5f;
  o[2] = hix - lox;
  o[3] = hiy - loy;
}

__global__ __launch_bounds__(256) void temporal_kernel(
    const float* __restrict__ tw, const float* __restrict__ shortcut,
    float* __restrict__ xi2)
{
  const int i = blockIdx.x * blockDim.x + threadIdx.x;
  if (i >= B_ * Q_ * C_) return;
  const int c  = i % C_;
  const int bq = i / C_;
  const int q  = bq % Q_;
  const int b  = bq / Q_;
  const float t0 = tw[(b * T_ + 0) * Q_ + q];
  const float t1 = tw[(b * T_ + 1) * Q_ + q];
  const float t2 = tw[(b * T_ + 2) * Q_ + q];
  const float m  = fmaxf(t0, fmaxf(t1, t2));
  const float e0 = __expf(t0 - m), e1 = __expf(t1 - m), e2 = __expf(t2 - m);
  const float inv = 1.f / (e0 + e1 + e2);
  const float v =
      e0 * shortcut[((size_t)(b * T_ + 0) * Q_ + q) * C_ + c] +
      e1 * shortcut[((size_t)(b * T_ + 1) * Q_ + q) * C_ + c] +
      e2 * shortcut[((size_t)(b * T_ + 2) * Q_ + q) * C_ + c];
  xi2[i] = v * inv;
}

// ---------------------------------------------------------------------------
// Host orchestration
// ---------------------------------------------------------------------------
static void gemm(hipStream_t s, const float* X, const float* W, const float* b,
                 float* Y, int M, int N, int K, int act)
{
  const int waves  = ((M + 31) / 32) * ((N + 63) / 64);
  const int blocks = (waves + 7) / 8;
  if (act == ACT_RELU)
    gemm_bias_kernel<ACT_RELU><<<blocks, 256, 0, s>>>(X, W, b, Y, M, N, K);
  else if (act == ACT_GELU)
    gemm_bias_kernel<ACT_GELU><<<blocks, 256, 0, s>>>(X, W, b, Y, M, N, K);
  else
    gemm_bias_kernel<ACT_NONE><<<blocks, 256, 0, s>>>(X, W, b, Y, M, N, K);
}

static void layernorm(hipStream_t s, const float* A, const float* Bres,
                      const float* g, const float* be, float* Y, int M)
{
  ln_kernel<<<(M + 7) / 8, 256, 0, s>>>(A, Bres, g, be, Y, M);
}

extern "C" void kernel_launch(void* const* d_in, const int* in_sizes, int n_in,
                              void* d_out, int out_size, void* d_ws, size_t ws_size,
                              hipStream_t stream)
{
  (void)in_sizes; (void)n_in; (void)out_size; (void)ws_size;

  const float* x_in   = (const float*)d_in[0];   // (BT,Q,C)
  const float* refpts = (const float*)d_in[1];   // (BT,Q,2)
  const float* srcs   = (const float*)d_in[2];   // (BT,N,C)
  // d_in[3] = spatial_shapes (hardcoded constants)

  // --- params: jax pytree flatten, dict keys sorted ---
  // top: bbox(6) | layers(3*38) | norm(2) | p2p(4)
  auto PRM = [&](int i) { return (const float*)d_in[4 + i]; };
  // layer leaf layout (38): ca{att W,b; off W,b; out W,b; val W,b}(8)
  //   ffn{l1 W,b; l2 W,b}(4) ffn_i(4)
  //   norm1 w,b; norm1_i; norm2; norm2_i; norm3; norm3_i (12)
  //   sa{Wi,Wo,bi,bo}(4) sa_i(4) tw{W,b}(2)
  auto Lp = [&](int l, int j) { return PRM(6 + l * 38 + j); };
  const float* BBW[3] = {PRM(0), PRM(2), PRM(4)};
  const float* BBB[3] = {PRM(1), PRM(3), PRM(5)};
  const float* NORM_w = PRM(120);
  const float* NORM_b = PRM(121);
  const float* P2P0_W = PRM(122);
  const float* P2P0_b = PRM(123);
  const float* P2P1_W = PRM(124);
  const float* P2P1_b = PRM(125);

  // --- workspace carve (~168 MB of fp32; activations stay resident in L2) ---
  float* Wb = (float*)d_ws;
  size_t cur = 0;
  auto take = [&](size_t n) { float* p = Wb + cur; cur += (n + 63) & ~(size_t)63; return p; };
  const size_t M1C = (size_t)M1_ * C_;
  float* valb  = take((size_t)BT_ * NSRC_ * C_);
  float* bigA  = take((size_t)M1_ * DFF_);
  float* xbuf  = take(M1C);
  float* xpos  = take(M1C);
  float* xq    = take(M1C);
  float* shcut = take(M1C);
  float* samp  = take(M1C);
  float* tA    = take(M1C);
  float* tB    = take(M1C);
  float* tC    = take(M1C);
  float* qb    = take(M1C);
  float* kb    = take(M1C);
  float* vb    = take(M1C);
  float* S     = take((size_t)BT_ * H_ * Q_ * Q_);
  float* offb  = take((size_t)M1_ * H_ * L_ * P_ * 2);
  float* attl  = take((size_t)M1_ * H_ * L_ * P_);
  float* xinst = take((size_t)M2_ * C_);
  float* xiq   = take((size_t)M2_ * C_);
  float* xi2   = take((size_t)M2_ * C_);
  float* ipos  = take((size_t)M2_ * C_);
  float* boxA  = take((size_t)M1_ * 4);
  float* boxB  = take((size_t)M1_ * 4);
  float* ref0b = take((size_t)M1_ * 4);
  float* bbout = take((size_t)M1_ * 4);
  float* irefb = take((size_t)M2_ * 4);
  float* pts4  = take((size_t)M1_ * 4);
  float* p2p0w4 = take((size_t)C_ * 4);
  float* twb   = take((size_t)M1_);

  float* OUT = (float*)d_out;
  float* oX = OUT;                              // (4, BT*Q, C)
  float* oI = oX + (size_t)4 * M1_ * C_;        // (4, B*Q, C)
  float* oB = oI + (size_t)4 * M2_ * C_;        // (4, BT*Q, 4)

  const float scale = 0.1767766953f;  // 1/sqrt(32)

  // Widen P2P0_W (256 rows, K=2) to K=4 with zero padding (pad columns always
  // multiply zero-padded input columns, so they never contribute).
  pad2to4_kernel<<<(C_ + 255) / 256, 256, 0, stream>>>(P2P0_W, p2p0w4, 0.f, C_);
  P2P0_W = p2p0w4;

  auto p2p = [&](const float* boxes4, float* dst, int M) {
    slice2pad4_kernel<<<(M + 255) / 256, 256, 0, stream>>>(boxes4, pts4, M);
    gemm(stream, pts4, P2P0_W, P2P0_b, tA, M, C_, 4, ACT_RELU);
    gemm(stream, tA, P2P1_W, P2P1_b, dst, M, C_, C_, ACT_NONE);
  };
  auto bbox_head = [&](const float* xin) {
    layernorm(stream, xin, nullptr, NORM_w, NORM_b, tA, M1_);
    gemm(stream, tA, BBW[0], BBB[0], tB, M1_, C_, C_, ACT_RELU);
    gemm(stream, tB, BBW[1], BBB[1], tC, M1_, C_, C_, ACT_RELU);
    gemm(stream, tC, BBW[2], BBB[2], bbout, M1_, 4, C_, ACT_NONE);
  };
  auto mha = [&](const float* q_in, const float* v_in, const float* Wi,
                 const float* bi, const float* Wo, const float* bo,
                 float* out, int Bn) {
    const int M = Bn * Q_;
    gemm(stream, q_in, Wi,            bi,            qb, M, C_, C_, ACT_NONE);
    gemm(stream, q_in, Wi + C_ * C_,  bi + C_,       kb, M, C_, C_, ACT_NONE);
    gemm(stream, v_in, Wi + 2*C_*C_,  bi + 2 * C_,   vb, M, C_, C_, ACT_NONE);
    const int tiles = (Q_ + 15) / 16;
    {
      const int waves = Bn * H_ * tiles * tiles;
      attn_scores_kernel<<<(waves + 7) / 8, 256, 0, stream>>>(qb, kb, S, Bn, Q_, scale);
    }
    softmax_rows_kernel<<<(Bn * H_ * Q_ + 7) / 8, 256, 0, stream>>>(S, Bn * H_ * Q_, Q_);
    {
      const int waves = Bn * H_ * tiles;
      attn_apply_kernel<<<(waves + 7) / 8, 256, 0, stream>>>(S, vb, tA, Bn, Q_);
    }
    gemm(stream, tA, Wo, bo, out, M, C_, C_, ACT_NONE);
  };

  // ------------------------------ prologue --------------------------------
  hipMemcpyAsync(xbuf, x_in, M1C * sizeof(float), hipMemcpyDeviceToDevice, stream);
  hipMemcpyAsync(oX, x_in, M1C * sizeof(float), hipMemcpyDeviceToDevice, stream);
  for (int b = 0; b < B_; ++b) {
    hipMemcpyAsync(xinst + (size_t)b * Q_ * C_,
                   x_in + (size_t)(b * T_ + CT_) * Q_ * C_,
                   (size_t)Q_ * C_ * sizeof(float), hipMemcpyDeviceToDevice, stream);
    hipMemcpyAsync(oI + (size_t)b * Q_ * C_,
                   x_in + (size_t)(b * T_ + CT_) * Q_ * C_,
                   (size_t)Q_ * C_ * sizeof(float), hipMemcpyDeviceToDevice, stream);
  }
  pad2to4_kernel<<<(M1_ + 255) / 256, 256, 0, stream>>>(refpts, ref0b, 0.1f, M1_);
  bbox_head(xbuf);
  sigboxes_kernel<<<(M1_ * 4 + 255) / 256, 256, 0, stream>>>(bbout, ref0b, boxA, M1_ * 4);
  float* ref = boxA;
  float* nxt = boxB;
  hipMemcpyAsync(oB, ref, (size_t)M1_ * 4 * sizeof(float), hipMemcpyDeviceToDevice, stream);
  p2p(ref, xpos, M1_);
  instboxes_kernel<<<(M2_ + 255) / 256, 256, 0, stream>>>(ref, irefb);
  p2p(irefb, ipos, M2_);

  // ------------------------------- layers ---------------------------------
  for (int l = 0; l < 3; ++l) {
    // ---- deformable cross-attention ----
    add_kernel<<<((int)M1C + 255) / 256, 256, 0, stream>>>(xbuf, xpos, xq, (int)M1C);
    gemm(stream, xq, Lp(l, 2), Lp(l, 3), offb, M1_, H_ * L_ * P_ * 2, C_, ACT_NONE);
    gemm(stream, xq, Lp(l, 0), Lp(l, 1), attl, M1_, H_ * L_ * P_, C_, ACT_NONE);
    gemm(stream, srcs, Lp(l, 6), Lp(l, 7), valb, BT_ * NSRC_, C_, C_, ACT_NONE);
    {
      const int waves = BT_ * Q_ * H_;
      msdeform_kernel<<<(waves + 7) / 8, 256, 0, stream>>>(valb, offb, attl, ref, samp);
    }
    gemm(stream, samp, Lp(l, 4), Lp(l, 5), tB, M1_, C_, C_, ACT_NONE);
    layernorm(stream, xbuf, tB, Lp(l, 20), Lp(l, 21), xbuf, M1_);  // norm2
    hipMemcpyAsync(shcut, xbuf, M1C * sizeof(float), hipMemcpyDeviceToDevice, stream);

    // ---- self attention ----
    add_kernel<<<((int)M1C + 255) / 256, 256, 0, stream>>>(xbuf, xpos, xq, (int)M1C);
    mha(xq, xbuf, Lp(l, 28), Lp(l, 30), Lp(l, 29), Lp(l, 31), tB, BT_);
    layernorm(stream, xbuf, tB, Lp(l, 16), Lp(l, 17), xbuf, M1_);  // norm1

    // ---- FFN ----
    gemm(stream, xbuf, Lp(l, 8), Lp(l, 9), bigA, M1_, DFF_, C_, ACT_GELU);
    gemm(stream, bigA, Lp(l, 10), Lp(l, 11), tB, M1_, C_, DFF_, ACT_NONE);
    layernorm(stream, xbuf, tB, Lp(l, 24), Lp(l, 25), xbuf, M1_);  // norm3

    // ---- temporal aggregation -> instance branch ----
    gemm(stream, xbuf, Lp(l, 36), Lp(l, 37), twb, M1_, 1, C_, ACT_NONE);
    temporal_kernel<<<(M2_ * C_ + 255) / 256, 256, 0, stream>>>(twb, shcut, xi2);
    layernorm(stream, xinst, xi2, Lp(l, 22), Lp(l, 23), xinst, M2_);  // norm2_i

    add_kernel<<<(M2_ * C_ + 255) / 256, 256, 0, stream>>>(xinst, ipos, xiq, M2_ * C_);
    mha(xiq, xinst, Lp(l, 32), Lp(l, 34), Lp(l, 33), Lp(l, 35), tB, B_);
    layernorm(stream, xinst, tB, Lp(l, 18), Lp(l, 19), xinst, M2_);  // norm1_i

    gemm(stream, xinst, Lp(l, 12), Lp(l, 13), bigA, M2_, DFF_, C_, ACT_GELU);
    gemm(stream, bigA, Lp(l, 14), Lp(l, 15), tB, M2_, C_, DFF_, ACT_NONE);
    layernorm(stream, xinst, tB, Lp(l, 26), Lp(l, 27), xinst, M2_);  // norm3_i

    // ---- box refinement + positional embeds for next layer ----
    bbox_head(xbuf);
    sigboxes_kernel<<<(M1_ * 4 + 255) / 256, 256, 0, stream>>>(bbout, ref, nxt, M1_ * 4);
    { float* t = ref; ref = nxt; nxt = t; }
    p2p(ref, xpos, M1_);
    instboxes_kernel<<<(M2_ + 255) / 256, 256, 0, stream>>>(ref, irefb);
    p2p(irefb, ipos, M2_);

    // ---- record intermediate outputs ----
    hipMemcpyAsync(oX + (size_t)(l + 1) * M1_ * C_, xbuf, M1C * sizeof(float),
                   hipMemcpyDeviceToDevice, stream);
    hipMemcpyAsync(oI + (size_t)(l + 1) * M2_ * C_, xinst,
                   (size_t)M2_ * C_ * sizeof(float), hipMemcpyDeviceToDevice, stream);
    hipMemcpyAsync(oB + (size_t)(l + 1) * M1_ * 4, ref,
                   (size_t)M1_ * 4 * sizeof(float), hipMemcpyDeviceToDevice, stream);
  }
}